// SpectralConv2d_472446403181
// MI455X (gfx1250) — compile-verified
//
#include <hip/hip_runtime.h>
#include <hip/hip_bf16.h>

// ---------------------------------------------------------------------------
// SpectralConv2d (FNO layer), B=16, CI=CO=64, H=W=256, M1=M2=16.
// Strategy: the output spectrum is zero except two 16x16 corner blocks, so we
// replace full FFTs with 5 dense-GEMM stages, all mapped onto
// v_wmma_f32_16x16x4_f32 (CDNA5 f32 matrix pipe, wave32).
// ---------------------------------------------------------------------------

typedef __attribute__((ext_vector_type(2))) float v2f;
typedef __attribute__((ext_vector_type(8))) float v8f;

// Problem dims
#define Bz   16
#define CIz  64
#define COz  64
#define Hz   256
#define Wz   256
#define M1z  16
#define M2z  16

// Workspace layout (in floats)
static constexpr size_t OFF_CWR  = 0;                          // [256][16] fwd W-DFT cos
static constexpr size_t OFF_CWI  = OFF_CWR + 256 * 16;         // [256][16] fwd W-DFT -sin
static constexpr size_t OFF_EHR  = OFF_CWI + 256 * 16;         // [32][256] fwd H-DFT (A-matrix)
static constexpr size_t OFF_EHI  = OFF_EHR + 32 * 256;
static constexpr size_t OFF_E2R  = OFF_EHI + 32 * 256;         // [256][32] inv H-DFT (A-matrix)
static constexpr size_t OFF_E2I  = OFF_E2R + 256 * 32;
static constexpr size_t OFF_CINV = OFF_E2I + 256 * 32;         // [32][256] inv W-DFT (B-matrix, re/im stacked)
static constexpr size_t OFF_X1R  = OFF_CINV + 32 * 256;        // [B*CI*H][16]
static constexpr size_t OFF_X1I  = OFF_X1R + (size_t)262144 * 16;
static constexpr size_t OFF_TR   = OFF_X1I + (size_t)262144 * 16;   // [512 modes][B*CI]
static constexpr size_t OFF_TI   = OFF_TR + (size_t)512 * 1024;
static constexpr size_t OFF_WTR  = OFF_TI + (size_t)512 * 1024;     // [512 modes][CI*CO]
static constexpr size_t OFF_WTI  = OFF_WTR + (size_t)512 * 4096;
static constexpr size_t OFF_YR   = OFF_WTI + (size_t)512 * 4096;    // [B*CO][512 modes]
static constexpr size_t OFF_YI   = OFF_YR + (size_t)1024 * 512;
static constexpr size_t OFF_ZR   = OFF_YI + (size_t)1024 * 512;     // [B*CO*H][16]
static constexpr size_t OFF_ZI   = OFF_ZR + (size_t)262144 * 16;
// total = OFF_ZI + 262144*16 floats  (~92.5 MB)

// ---------------------------------------------------------------------------
// WMMA f32 16x16x4 fragment helpers (ISA 7.12.2 layouts, wave32)
// ---------------------------------------------------------------------------
__device__ __forceinline__ v8f wmma4(v2f a, v2f b, v8f c) {
  // 8 args: (neg_a, A, neg_b, B, c_mod, C, reuse_a, reuse_b)
  return __builtin_amdgcn_wmma_f32_16x16x4_f32(false, a, false, b, (short)0, c,
                                               false, false);
}

// A fragment: A[M=lane&15][K = kk + 2*(lane>=16) + {0,1}], row stride ld
__device__ __forceinline__ v2f frag_a(const float* __restrict__ p, int row0,
                                      int ld, int kk, int lane) {
  int m = lane & 15;
  int k = kk + ((lane >> 4) << 1);
  const float* q = p + (size_t)(row0 + m) * ld + k;
  v2f a;
  a.x = q[0];
  a.y = q[1];
  return a;
}

__device__ __forceinline__ v2f frag_a_nt(const float* __restrict__ p, int row0,
                                         int ld, int kk, int lane) {
  int m = lane & 15;
  int k = kk + ((lane >> 4) << 1);
  const float* q = p + (size_t)(row0 + m) * ld + k;
  v2f a;
  a.x = __builtin_nontemporal_load(q);
  a.y = __builtin_nontemporal_load(q + 1);
  return a;
}

// B fragment: B[K = kk + 2*(lane>=16) + {0,1}][N = n0 + (lane&15)], row stride ld
__device__ __forceinline__ v2f frag_b(const float* __restrict__ p, int kk,
                                      int ld, int n0, int lane) {
  int n = n0 + (lane & 15);
  int k = kk + ((lane >> 4) << 1);
  v2f b;
  b.x = p[(size_t)k * ld + n];
  b.y = p[(size_t)(k + 1) * ld + n];
  return b;
}

// C/D fragment store: VGPR r holds M=r (lanes 0-15) and M=r+8 (lanes 16-31)
__device__ __forceinline__ void frag_store(float* __restrict__ p, v8f c,
                                           int row0, int ld, int n0, int lane) {
  int n = n0 + (lane & 15);
  int roff = (lane >> 4) << 3;  // 0 or 8
#pragma unroll
  for (int r = 0; r < 8; ++r)
    p[(size_t)(row0 + r + roff) * ld + n] = c[r];
}

__device__ __forceinline__ void frag_store_nt(float* __restrict__ p, v8f c,
                                              int row0, int ld, int n0, int lane) {
  int n = n0 + (lane & 15);
  int roff = (lane >> 4) << 3;
#pragma unroll
  for (int r = 0; r < 8; ++r)
    __builtin_nontemporal_store(c[r], p + (size_t)(row0 + r + roff) * ld + n);
}

// ---------------------------------------------------------------------------
// Stage 0a: build all DFT twiddle tables in workspace (ortho norm = 1/16 per stage)
// ---------------------------------------------------------------------------
__global__ void k_twiddle(float* __restrict__ ws) {
  const float PI2 = 6.28318530717958647692f;
  int i = blockIdx.x * blockDim.x + threadIdx.x;  // [0, 28672)
  if (i < 4096) {
    // Cw: [w][kx], forward W-DFT (B-matrix of stage 1)
    int w = i >> 4, kx = i & 15;
    float ang = PI2 * (float)((w * kx) & 255) * (1.0f / 256.0f);
    ws[OFF_CWR + i] = __cosf(ang) * (1.0f / 16.0f);
    ws[OFF_CWI + i] = -__sinf(ang) * (1.0f / 16.0f);
  } else if (i < 12288) {
    // Eh: [m][h], m<16 -> ky=m ; m>=16 -> ky=224+m (i.e. 240..255)
    int j = i - 4096;
    int m = j >> 8, h = j & 255;
    int ky = (m < 16) ? m : (224 + m);
    float ang = PI2 * (float)((ky * h) & 255) * (1.0f / 256.0f);
    ws[OFF_EHR + j] = __cosf(ang) * (1.0f / 16.0f);
    ws[OFF_EHI + j] = -__sinf(ang) * (1.0f / 16.0f);
  } else if (i < 20480) {
    // E2: [h][m], inverse H-DFT (positive exponent)
    int j = i - 12288;
    int h = j >> 5, m = j & 31;
    int ky = (m < 16) ? m : (224 + m);
    float ang = PI2 * (float)((ky * h) & 255) * (1.0f / 256.0f);
    ws[OFF_E2R + j] = __cosf(ang) * (1.0f / 16.0f);
    ws[OFF_E2I + j] = __sinf(ang) * (1.0f / 16.0f);
  } else {
    // Cinv: [k][w]; rows 0..15 multiply ZR, rows 16..31 multiply ZI.
    // irfft symmetry: kx=0 weight 1, kx=1..15 weight 2 (halves are conjugates).
    int j = i - 20480;
    int k = j >> 8, w = j & 255;
    int kx = k & 15;
    float ang = PI2 * (float)((kx * w) & 255) * (1.0f / 256.0f);
    float cf = ((kx == 0) ? 1.0f : 2.0f) * (1.0f / 16.0f);
    ws[OFF_CINV + j] = (k < 16) ? cf * __cosf(ang) : -cf * __sinf(ang);
  }
}

// ---------------------------------------------------------------------------
// Stage 0b: transpose weights (CI,CO,16,16,2) -> Wt[mode q][i*64+o] (re/im)
// q = m*16+kx ; m<16 -> w1[ky=m], m>=16 -> w2[ky=m-16]
// ---------------------------------------------------------------------------
__global__ void k_wt(const float* __restrict__ w1, const float* __restrict__ w2,
                     float* __restrict__ WtR, float* __restrict__ WtI) {
  int idx = blockIdx.x * blockDim.x + threadIdx.x;  // [0, 512*4096)
  int q = idx >> 12;
  int io = idx & 4095;
  int i = io >> 6, o = io & 63;
  int m = q >> 4, kx = q & 15;
  const float* w = (m < 16) ? w1 : w2;
  int ky = (m < 16) ? m : (m - 16);
  size_t src = ((((size_t)i * 64 + o) * 16 + ky) * 16 + kx) * 2;
  WtR[idx] = w[src];
  WtI[idx] = w[src + 1];
}

// ---------------------------------------------------------------------------
// Stage 1: forward W-DFT.  X1[row][kx] = sum_w x[row][w] * Cw[w][kx]
// rows = B*CI*H = 262144, one 16-row tile per wave; K=256, N=16(re)+16(im).
// ---------------------------------------------------------------------------
__global__ void k_fwd_w(const float* __restrict__ x, const float* __restrict__ ws,
                        float* __restrict__ X1R, float* __restrict__ X1I) {
  int lane = threadIdx.x & 31;
  int wv = blockIdx.x * (blockDim.x >> 5) + (threadIdx.x >> 5);  // [0, 16384)
  int row0 = wv << 4;
  const float* CwR = ws + OFF_CWR;
  const float* CwI = ws + OFF_CWI;
  v8f cr = {}, ci = {};
#pragma unroll 4
  for (int kk = 0; kk < 256; kk += 4) {
    v2f a = frag_a_nt(x, row0, 256, kk, lane);  // stream input, don't pollute L2
    v2f br = frag_b(CwR, kk, 16, 0, lane);
    v2f bi = frag_b(CwI, kk, 16, 0, lane);
    cr = wmma4(a, br, cr);
    ci = wmma4(a, bi, ci);
  }
  frag_store(X1R, cr, row0, 16, 0, lane);
  frag_store(X1I, ci, row0, 16, 0, lane);
}

// ---------------------------------------------------------------------------
// Stage 2: forward H-DFT (complex).  T[m][kx] = sum_h Eh[m][h] * X1[h][kx]
// One wave per (bi, ky-tile); K=256. Output scattered mode-major for stage 3.
// ---------------------------------------------------------------------------
__global__ void k_fwd_h(const float* __restrict__ ws, float* __restrict__ TR,
                        float* __restrict__ TI) {
  int lane = threadIdx.x & 31;
  int wv = blockIdx.x * (blockDim.x >> 5) + (threadIdx.x >> 5);  // [0, 2048)
  int bi = wv >> 1;
  int t = (wv & 1) << 4;  // ky-row tile: 0 or 16
  const float* EhR = ws + OFF_EHR;
  const float* EhI = ws + OFF_EHI;
  const float* bR = ws + OFF_X1R + (size_t)bi * 4096;
  const float* bI = ws + OFF_X1I + (size_t)bi * 4096;
  v8f cr = {}, ci = {};
#pragma unroll 4
  for (int kk = 0; kk < 256; kk += 4) {
    v2f ar = frag_a(EhR, t, 256, kk, lane);
    v2f ai = frag_a(EhI, t, 256, kk, lane);
    v2f br = frag_b(bR, kk, 16, 0, lane);
    v2f bi2 = frag_b(bI, kk, 16, 0, lane);
    v2f ain = -ai;  // f32 WMMA has no A-negate modifier (ISA 7.12 NEG table)
    cr = wmma4(ar, br, cr);
    cr = wmma4(ain, bi2, cr);
    ci = wmma4(ar, bi2, ci);
    ci = wmma4(ai, br, ci);
  }
  int n = lane & 15;
  int roff = (lane >> 4) << 3;
#pragma unroll
  for (int r = 0; r < 8; ++r) {
    int m = t + r + roff;
    TR[(size_t)(m * 16 + n) * 1024 + bi] = cr[r];
    TI[(size_t)(m * 16 + n) * 1024 + bi] = ci[r];
  }
}

// ---------------------------------------------------------------------------
// Stage 3: per-mode channel mix (complex).  Y[b][o] = sum_i T[b][i] * Wt[i][o]
// One wave per (mode q, CO-tile); M=16(b), K=64(i), N=16 of 64(o).
// ---------------------------------------------------------------------------
__global__ void k_mix(const float* __restrict__ ws_t, float* __restrict__ YR,
                      float* __restrict__ YI) {
  int lane = threadIdx.x & 31;
  int wv = blockIdx.x * (blockDim.x >> 5) + (threadIdx.x >> 5);  // [0, 2048)
  int q = wv >> 2;
  int n0 = (wv & 3) << 4;
  const float* aR = ws_t + OFF_TR + (size_t)q * 1024;   // [b][i] ld=64
  const float* aI = ws_t + OFF_TI + (size_t)q * 1024;
  const float* bR = ws_t + OFF_WTR + (size_t)q * 4096;  // [i][o] ld=64
  const float* bI = ws_t + OFF_WTI + (size_t)q * 4096;
  v8f cr = {}, ci = {};
#pragma unroll
  for (int kk = 0; kk < 64; kk += 4) {
    v2f ar = frag_a(aR, 0, 64, kk, lane);
    v2f ai = frag_a(aI, 0, 64, kk, lane);
    v2f br = frag_b(bR, kk, 64, n0, lane);
    v2f bi2 = frag_b(bI, kk, 64, n0, lane);
    v2f ain = -ai;
    cr = wmma4(ar, br, cr);
    cr = wmma4(ain, bi2, cr);
    ci = wmma4(ar, bi2, ci);
    ci = wmma4(ai, br, ci);
  }
  int n = lane & 15;
  int roff = (lane >> 4) << 3;
#pragma unroll
  for (int r = 0; r < 8; ++r) {
    int b = r + roff;
    int o = n0 + n;
    YR[(size_t)(b * 64 + o) * 512 + q] = cr[r];
    YI[(size_t)(b * 64 + o) * 512 + q] = ci[r];
  }
}

// ---------------------------------------------------------------------------
// Stage 4: inverse H-DFT (complex).  Z[h][kx] = sum_m E2[h][m] * Y[m][kx]
// One wave per (bo, h-tile); K=32, N=16.
// ---------------------------------------------------------------------------
__global__ void k_inv_h(const float* __restrict__ ws, float* __restrict__ ZR,
                        float* __restrict__ ZI) {
  int lane = threadIdx.x & 31;
  int wv = blockIdx.x * (blockDim.x >> 5) + (threadIdx.x >> 5);  // [0, 16384)
  int bo = wv >> 4;
  int h0 = (wv & 15) << 4;
  const float* E2R = ws + OFF_E2R;
  const float* E2I = ws + OFF_E2I;
  const float* bR = ws + OFF_YR + (size_t)bo * 512;  // [m][kx] ld=16
  const float* bI = ws + OFF_YI + (size_t)bo * 512;
  v8f cr = {}, ci = {};
#pragma unroll
  for (int kk = 0; kk < 32; kk += 4) {
    v2f ar = frag_a(E2R, h0, 32, kk, lane);
    v2f ai = frag_a(E2I, h0, 32, kk, lane);
    v2f br = frag_b(bR, kk, 16, 0, lane);
    v2f bi2 = frag_b(bI, kk, 16, 0, lane);
    v2f ain = -ai;
    cr = wmma4(ar, br, cr);
    cr = wmma4(ain, bi2, cr);
    ci = wmma4(ar, bi2, ci);
    ci = wmma4(ai, br, ci);
  }
  int row0 = bo * 256 + h0;
  frag_store(ZR, cr, row0, 16, 0, lane);
  frag_store(ZI, ci, row0, 16, 0, lane);
}

// ---------------------------------------------------------------------------
// Stage 5: inverse W-DFT, real output.
// out[row][w] = sum_kx ZR[row][kx]*Cinv[kx][w] + ZI[row][kx]*Cinv[16+kx][w]
// One wave per (row-tile, w-tile); K=16(re)+16(im), nontemporal 268 MB store.
// ---------------------------------------------------------------------------
__global__ void k_inv_w(const float* __restrict__ ws, float* __restrict__ out) {
  int lane = threadIdx.x & 31;
  int wv = blockIdx.x * (blockDim.x >> 5) + (threadIdx.x >> 5);  // [0, 262144)
  int row0 = (wv >> 4) << 4;
  int n0 = (wv & 15) << 4;
  const float* ZR = ws + OFF_ZR;
  const float* ZI = ws + OFF_ZI;
  const float* Cinv = ws + OFF_CINV;  // [32][256]
  v8f c = {};
#pragma unroll
  for (int kk = 0; kk < 16; kk += 4) {
    v2f ar = frag_a(ZR, row0, 16, kk, lane);
    v2f b1 = frag_b(Cinv, kk, 256, n0, lane);
    c = wmma4(ar, b1, c);
    v2f ai = frag_a(ZI, row0, 16, kk, lane);
    v2f b2 = frag_b(Cinv + 16 * 256, kk, 256, n0, lane);
    c = wmma4(ai, b2, c);
  }
  frag_store_nt(out, c, row0, 256, n0, lane);
}

// ---------------------------------------------------------------------------
extern "C" void kernel_launch(void* const* d_in, const int* in_sizes, int n_in,
                              void* d_out, int out_size, void* d_ws,
                              size_t ws_size, hipStream_t stream) {
  const float* x = (const float*)d_in[0];   // (16,64,256,256)
  const float* w1 = (const float*)d_in[1];  // (64,64,16,16,2)
  const float* w2 = (const float*)d_in[2];  // (64,64,16,16,2)
  float* out = (float*)d_out;               // (16,64,256,256)
  float* ws = (float*)d_ws;

  // Stage 0: twiddle tables (28672 threads) + weight transpose (2M threads)
  k_twiddle<<<112, 256, 0, stream>>>(ws);
  k_wt<<<8192, 256, 0, stream>>>(w1, w2, ws + OFF_WTR, ws + OFF_WTI);

  // Stage 1: 16384 row-tiles, 8 waves/block
  k_fwd_w<<<2048, 256, 0, stream>>>(x, ws, ws + OFF_X1R, ws + OFF_X1I);
  // Stage 2: 2048 wave-tasks
  k_fwd_h<<<256, 256, 0, stream>>>(ws, ws + OFF_TR, ws + OFF_TI);
  // Stage 3: 2048 wave-tasks
  k_mix<<<256, 256, 0, stream>>>(ws, ws + OFF_YR, ws + OFF_YI);
  // Stage 4: 16384 wave-tasks
  k_inv_h<<<2048, 256, 0, stream>>>(ws, ws + OFF_ZR, ws + OFF_ZI);
  // Stage 5: 262144 wave-tasks
  k_inv_w<<<32768, 256, 0, stream>>>(ws, out);
}